// BiRGAT_25692494364912
// MI455X (gfx1250) — compile-verified
//
#include <hip/hip_runtime.h>
#include <hip/hip_bf16.h>

typedef __attribute__((ext_vector_type(2))) float v2f;
typedef __attribute__((ext_vector_type(8))) float v8f;

#define N_S 4000
#define N_G 50000
#define NEDGE 250000
#define NHEAD 4

// ---------------- utility fills ----------------
__global__ void fill_f32_kernel(float* p, float v, int n) {
    int i = blockIdx.x * blockDim.x + threadIdx.x;
    if (i < n) p[i] = v;
}
__global__ void fill_u32_kernel(unsigned* p, unsigned v, int n) {
    int i = blockIdx.x * blockDim.x + threadIdx.x;
    if (i < n) p[i] = v;
}

// ---------------- WMMA fp32 GEMM: C[M,N] = A[M,K] @ W[K,N] + bias[N] ----------------
// One wave per 16 x (16*NT) output strip: A fragment reused across NT independent
// accumulator chains (better XDL overlap than one serial chain).
// A 16x4 layout: lanes 0-15 rows M=0..15 (K=kh,kh+1), lanes 16-31 same rows (K=kh+2,kh+3).
// B 4x16 layout: row striped across lanes; C/D: 8 VGPRs, lanes 16-31 hold M+8.
template <int NT>
__global__ __launch_bounds__(256) void wmma_gemm_bias_kernel(
    const float* __restrict__ A, const float* __restrict__ W,
    const float* __restrict__ bias, float* __restrict__ C,
    int M, int N, int K) {
    int wave = (blockIdx.x * blockDim.x + threadIdx.x) >> 5;
    int lane = threadIdx.x & 31;
    int tilesM = M >> 4;
    int groupsN = N / (16 * NT);
    int tm = wave / groupsN;
    int tg = wave - tm * groupsN;
    if (tm >= tilesM) return;            // wave-uniform: EXEC stays all-ones inside

    int row = lane & 15;
    int kh  = (lane >> 4) << 1;          // 0 for lanes 0-15, 2 for lanes 16-31
    v8f acc[NT];
#pragma unroll
    for (int t = 0; t < NT; ++t) acc[t] = (v8f){};

    const float* Ap = A + (size_t)(tm * 16 + row) * K;
    const float* Wp = W + (tg * 16 * NT + row);
    for (int k = 0; k < K; k += 4) {
        v2f a;
        a.x = Ap[k + kh];
        a.y = Ap[k + kh + 1];
        const float* Wk0 = Wp + (size_t)(k + kh) * N;
        const float* Wk1 = Wp + (size_t)(k + kh + 1) * N;
#pragma unroll
        for (int t = 0; t < NT; ++t) {
            v2f b;
            b.x = Wk0[t * 16];
            b.y = Wk1[t * 16];
            acc[t] = __builtin_amdgcn_wmma_f32_16x16x4_f32(false, a, false, b,
                                                           (short)0, acc[t], false, false);
        }
    }
    int crow = tm * 16 + (lane >> 4) * 8;
#pragma unroll
    for (int t = 0; t < NT; ++t) {
        int ccol = tg * 16 * NT + t * 16 + row;
        float bv = bias[ccol];
#pragma unroll
        for (int r = 0; r < 8; ++r)
            C[(size_t)(crow + r) * N + ccol] = acc[t][r] + bv;
    }
}

// ---------------- edge pass 1: logits + segment max ----------------
// warp per edge; each lane owns VEC=HC/32 contiguous channels via float4 loads
// (stays within one head for H=4); per-head reduce over LPH=32/H lanes.
template <int HC, int C>
__global__ __launch_bounds__(256) void edge_logits_kernel(
    const int* __restrict__ src, const int* __restrict__ dst,
    const float* __restrict__ xl, const float* __restrict__ xr,
    const float* __restrict__ ea, const float* __restrict__ We,
    const float* __restrict__ att,
    float* __restrict__ logits, unsigned* __restrict__ maxkey, int E) {
    constexpr int VEC = HC / 32;
    constexpr int H = HC / C;
    constexpr int LPH = 32 / H;
    int e = (blockIdx.x * blockDim.x + threadIdx.x) >> 5;
    int lane = threadIdx.x & 31;
    if (e >= E) return;
    int s = src[e], d = dst[e];
    float ev = ea[e];
    int ch0 = lane * VEC;
    const float4* pl = (const float4*)(xl + (size_t)s * HC + ch0);
    const float4* pr = (const float4*)(xr + (size_t)d * HC + ch0);
    const float4* pw = (const float4*)(We + ch0);
    const float4* pa = (const float4*)(att + ch0);
    float part = 0.f;
#pragma unroll
    for (int i = 0; i < VEC / 4; ++i) {
        float4 l = pl[i], r = pr[i], w = pw[i], a4 = pa[i];
        float m0 = l.x + r.x + ev * w.x;  m0 = (m0 > 0.f) ? m0 : 0.2f * m0;
        float m1 = l.y + r.y + ev * w.y;  m1 = (m1 > 0.f) ? m1 : 0.2f * m1;
        float m2 = l.z + r.z + ev * w.z;  m2 = (m2 > 0.f) ? m2 : 0.2f * m2;
        float m3 = l.w + r.w + ev * w.w;  m3 = (m3 > 0.f) ? m3 : 0.2f * m3;
        part += m0 * a4.x + m1 * a4.y + m2 * a4.z + m3 * a4.w;
    }
#pragma unroll
    for (int off = 1; off < LPH; off <<= 1)
        part += __shfl_xor(part, off, 32);
    if ((lane & (LPH - 1)) == 0) {
        int h = lane / LPH;
        logits[(size_t)e * H + h] = part;
        unsigned bits = __float_as_uint(part);
        unsigned key = (bits & 0x80000000u) ? ~bits : (bits | 0x80000000u);
        atomicMax(&maxkey[(size_t)d * H + h], key);
    }
}

// ---------------- edge pass 2: exp + segment sum ----------------
__global__ void edge_expsum_kernel(const int* __restrict__ dst,
                                   const float* __restrict__ logits,
                                   const unsigned* __restrict__ maxkey,
                                   float* __restrict__ ex, float* __restrict__ denom,
                                   int E, int H) {
    int i = blockIdx.x * blockDim.x + threadIdx.x;
    if (i >= E * H) return;
    int e = i / H, h = i - e * H;
    int d = dst[e];
    unsigned key = maxkey[(size_t)d * H + h];
    unsigned bits = (key & 0x80000000u) ? (key & 0x7FFFFFFFu) : ~key;
    float mx = __uint_as_float(bits);
    float v = __expf(logits[i] - mx);
    ex[i] = v;
    atomicAdd(&denom[(size_t)d * H + h], v);
}

// ---------------- edge pass 3: alpha-weighted scatter ----------------
template <int HC, int C>
__global__ __launch_bounds__(256) void edge_scatter_kernel(
    const int* __restrict__ src, const int* __restrict__ dst,
    const float* __restrict__ xl, const float* __restrict__ ex,
    const float* __restrict__ denom, float* __restrict__ agg, int E) {
    constexpr int VEC = HC / 32;
    constexpr int H = HC / C;
    int e = (blockIdx.x * blockDim.x + threadIdx.x) >> 5;
    int lane = threadIdx.x & 31;
    if (e >= E) return;
    int s = src[e], d = dst[e];
    int ch0 = lane * VEC;
    int h = ch0 / C;
    float alpha = ex[(size_t)e * H + h] / denom[(size_t)d * H + h];
    const float4* pl = (const float4*)(xl + (size_t)s * HC + ch0);
    float* pa = agg + (size_t)d * HC + ch0;
#pragma unroll
    for (int i = 0; i < VEC / 4; ++i) {
        float4 l = pl[i];
        atomicAdd(&pa[4 * i + 0], l.x * alpha);
        atomicAdd(&pa[4 * i + 1], l.y * alpha);
        atomicAdd(&pa[4 * i + 2], l.z * alpha);
        atomicAdd(&pa[4 * i + 3], l.w * alpha);
    }
}

// ---------------- finalizers ----------------
__global__ void relu_bias_kernel(const float* __restrict__ agg,
                                 const float* __restrict__ bias,
                                 float* __restrict__ out, int total, int HC) {
    int i = blockIdx.x * blockDim.x + threadIdx.x;
    if (i >= total) return;
    int c = i % HC;
    float v = agg[i] + bias[c];
    out[i] = v > 0.f ? v : 0.f;
}

__global__ void relu_bias_self_kernel(const float* __restrict__ agg,
                                      const float* __restrict__ bias,
                                      const float* __restrict__ sl,
                                      float* __restrict__ out,
                                      int total, int HC, int Cs) {
    int i = blockIdx.x * blockDim.x + threadIdx.x;
    if (i >= total) return;
    int n = i / HC, c = i - n * HC;
    float v = agg[i] + bias[c] + sl[(size_t)n * Cs + (c % Cs)];
    out[i] = v > 0.f ? v : 0.f;
}

__global__ void finalize3_kernel(const float* __restrict__ agg,
                                 const float* __restrict__ bias,
                                 const float* __restrict__ sl,
                                 float* __restrict__ out, int Nn, int H, int C) {
    int i = blockIdx.x * blockDim.x + threadIdx.x;
    if (i >= Nn * C) return;
    int n = i / C, c = i - n * C;
    float s = 0.f;
    for (int h = 0; h < H; ++h)
        s += agg[(size_t)n * H * C + h * C + c];
    float v = s * (1.0f / H) + bias[c] + sl[i];
    out[i] = v > 0.f ? v : 0.f;
}

// ---------------- host launch ----------------
extern "C" void kernel_launch(void* const* d_in, const int* in_sizes, int n_in,
                              void* d_out, int out_size, void* d_ws, size_t ws_size,
                              hipStream_t stream) {
    const float* x_mrna  = (const float*)d_in[0];
    const float* x_gene  = (const float*)d_in[1];
    const int*   sg_src  = (const int*)d_in[2];
    const int*   sg_dst  = (const int*)d_in[3];
    const int*   gs_src  = (const int*)d_in[4];
    const int*   gs_dst  = (const int*)d_in[5];
    const float* ea_sg   = (const float*)d_in[6];
    const float* ea_gs   = (const float*)d_in[7];
    const float* Wl1_sg = (const float*)d_in[8],  *bl1_sg = (const float*)d_in[9];
    const float* Wr1_sg = (const float*)d_in[10], *br1_sg = (const float*)d_in[11];
    const float* We1_sg = (const float*)d_in[12], *att1_sg = (const float*)d_in[13];
    const float* bo1_sg = (const float*)d_in[14];
    const float* Wl1_gs = (const float*)d_in[15], *bl1_gs = (const float*)d_in[16];
    const float* Wr1_gs = (const float*)d_in[17], *br1_gs = (const float*)d_in[18];
    const float* We1_gs = (const float*)d_in[19], *att1_gs = (const float*)d_in[20];
    const float* bo1_gs = (const float*)d_in[21];
    const float* Wl3_gs = (const float*)d_in[22], *bl3_gs = (const float*)d_in[23];
    const float* Wr3_gs = (const float*)d_in[24], *br3_gs = (const float*)d_in[25];
    const float* We3_gs = (const float*)d_in[26], *att3_gs = (const float*)d_in[27];
    const float* bo3_gs = (const float*)d_in[28];
    const float* Wsl1 = (const float*)d_in[29], *bsl1 = (const float*)d_in[30];
    const float* Wsl3 = (const float*)d_in[31], *bsl3 = (const float*)d_in[32];
    float* out = (float*)d_out;
    float* ws = (float*)d_ws;

    const int HC1 = 128, HC3 = 256, C1 = 32, C3 = 64;

    // workspace layout (floats), with deliberate aliasing for layer-3 projections
    const size_t o_xl1_sg = 0;                              // 512000
    const size_t o_xr1_sg = 512000;                         // 6400000
    const size_t o_xl1_gs = 6912000;                        // 6400000
    const size_t o_xr1_gs = 13312000;                       // 512000
    const size_t o_sl1    = 13824000;                       // 128000
    const size_t o_x1g    = 13952000;                       // 6400000
    const size_t o_x1m    = 20352000;                       // 512000
    const size_t o_sl3    = 20864000;                       // 256000
    const size_t o_logit  = 21120000;                       // 1000000
    const size_t o_ex     = 22120000;                       // 1000000
    const size_t o_key    = 23120000;                       // 200000 (as u32)
    const size_t o_den    = 23320000;                       // 200000
    const size_t o_agg    = 23520000;                       // 6400000 -> end 29920000
    const size_t o_xl3    = 0;         // alias dead xl1_sg/xr1_sg/xl1_gs (12.8M < 13.312M)
    const size_t o_xr3    = 13312000;  // alias dead xr1_gs/sl1/(start of dead x1g)

    float*    logits = ws + o_logit;
    float*    exb    = ws + o_ex;
    unsigned* keyb   = (unsigned*)(ws + o_key);
    float*    denb   = ws + o_den;
    float*    agg    = ws + o_agg;

    auto gemm = [&](const float* A, const float* W, const float* b, float* Cp,
                    int M, int N, int K) {
        if (N % 64 == 0) {
            long waves = (long)(M >> 4) * (N >> 6);
            int blocks = (int)((waves * 32 + 255) / 256);
            wmma_gemm_bias_kernel<4><<<blocks, 256, 0, stream>>>(A, W, b, Cp, M, N, K);
        } else {  // N == 32
            long waves = (long)(M >> 4) * (N >> 5);
            int blocks = (int)((waves * 32 + 255) / 256);
            wmma_gemm_bias_kernel<2><<<blocks, 256, 0, stream>>>(A, W, b, Cp, M, N, K);
        }
    };
    auto fillf = [&](float* p, float v, int n) {
        fill_f32_kernel<<<(n + 255) / 256, 256, 0, stream>>>(p, v, n);
    };
    auto edge_pipeline = [&](const int* src, const int* dst, const float* xl,
                             const float* xr, const float* ea, const float* We,
                             const float* att, int ndst, int HC, int C) {
        int H = HC / C;
        fill_u32_kernel<<<(ndst * H + 255) / 256, 256, 0, stream>>>(keyb, 0x007FFFFFu, ndst * H);
        fillf(denb, 0.f, ndst * H);
        fillf(agg, 0.f, ndst * HC);
        int eblocks = (NEDGE * 32 + 255) / 256;
        if (HC == 128) {
            edge_logits_kernel<128, 32><<<eblocks, 256, 0, stream>>>(src, dst, xl, xr, ea,
                                                                     We, att, logits, keyb, NEDGE);
        } else {
            edge_logits_kernel<256, 64><<<eblocks, 256, 0, stream>>>(src, dst, xl, xr, ea,
                                                                     We, att, logits, keyb, NEDGE);
        }
        edge_expsum_kernel<<<(NEDGE * H + 255) / 256, 256, 0, stream>>>(dst, logits, keyb,
                                                                        exb, denb, NEDGE, H);
        if (HC == 128) {
            edge_scatter_kernel<128, 32><<<eblocks, 256, 0, stream>>>(src, dst, xl, exb,
                                                                      denb, agg, NEDGE);
        } else {
            edge_scatter_kernel<256, 64><<<eblocks, 256, 0, stream>>>(src, dst, xl, exb,
                                                                      denb, agg, NEDGE);
        }
    };

    // ---- layer 1 projections ----
    gemm(x_mrna, Wl1_sg, bl1_sg, ws + o_xl1_sg, N_S, HC1, 128);
    gemm(x_gene, Wr1_sg, br1_sg, ws + o_xr1_sg, N_G, HC1, 128);
    gemm(x_gene, Wl1_gs, bl1_gs, ws + o_xl1_gs, N_G, HC1, 128);
    gemm(x_mrna, Wr1_gs, br1_gs, ws + o_xr1_gs, N_S, HC1, 128);
    gemm(x_mrna, Wsl1, bsl1, ws + o_sl1, N_S, C1, 128);

    // ---- relation mrna->gene (dst = gene) ----
    edge_pipeline(sg_src, sg_dst, ws + o_xl1_sg, ws + o_xr1_sg, ea_sg, We1_sg, att1_sg,
                  N_G, HC1, C1);
    relu_bias_kernel<<<(N_G * HC1 + 255) / 256, 256, 0, stream>>>(agg, bo1_sg, ws + o_x1g,
                                                                  N_G * HC1, HC1);

    // ---- relation gene->mrna (dst = mrna) ----
    edge_pipeline(gs_src, gs_dst, ws + o_xl1_gs, ws + o_xr1_gs, ea_gs, We1_gs, att1_gs,
                  N_S, HC1, C1);
    relu_bias_self_kernel<<<(N_S * HC1 + 255) / 256, 256, 0, stream>>>(
        agg, bo1_gs, ws + o_sl1, ws + o_x1m, N_S * HC1, HC1, C1);

    // ---- layer 3 projections (sl3 first; xl3 before xr3 because xr3 aliases dead x1g head) ----
    gemm(ws + o_x1m, Wsl3, bsl3, ws + o_sl3, N_S, C3, 128);
    gemm(ws + o_x1g, Wl3_gs, bl3_gs, ws + o_xl3, N_G, HC3, 128);
    gemm(ws + o_x1m, Wr3_gs, br3_gs, ws + o_xr3, N_S, HC3, 128);

    // ---- layer 3 relation gene->mrna (dst = mrna), head-mean output ----
    edge_pipeline(gs_src, gs_dst, ws + o_xl3, ws + o_xr3, ea_gs, We3_gs, att3_gs,
                  N_S, HC3, C3);
    finalize3_kernel<<<(N_S * C3 + 255) / 256, 256, 0, stream>>>(agg, bo3_gs, ws + o_sl3,
                                                                 out, N_S, NHEAD, C3);
}